// LiftSplatShoot_74311524156033
// MI455X (gfx1250) — compile-verified
//
#include <hip/hip_runtime.h>
#include <hip/hip_bf16.h>

// ---------------- problem constants ----------------
#define HH   128
#define WW   352
#define HWSZ (HH * WW)          // 45056
#define BB   2
#define NN   6
#define DD   48
#define BEVC 64
#define BNC  (BB * NN)          // 12
#define PCSIZE ((size_t)BNC * 3 * HWSZ * DD)   // 77,856,768 floats

// ---------------- workspace layout (float offsets) ----------------
#define OFF_KINV 0                         // 12 * 9 = 108
#define OFF_DA   128                       // 48
#define OFF_S1   192                       // 64
#define OFF_Q1   256                       // 64
#define OFF_S2   320                       // 64
#define OFF_Q2   384                       // 64
#define OFF_VA   512                       // NN*HWSZ = 270336 -> ends 270848
#define OFF_BEVF 270848                    // BB*3*HWSZ = 270336 -> ends 541184
#define OFF_H    541184                    // BB*64*HWSZ = 5767168 -> ends 6308352
#define OFF_Y2   6308352                   // BB*64*HWSZ = 5767168 -> ends 12075520
#define OFF_WT   12075520                  // 64*576 = 36864 -> ends 12112384
// total ws need: 12,112,384 floats = ~48.5 MB

typedef float v2f __attribute__((ext_vector_type(2)));
typedef float v8f __attribute__((ext_vector_type(8)));

// ---------------------------------------------------------------
// Kernel 1: tiny prep — 12 closed-form 3x3 intrinsic inverses and
// the depth-attention softmax (48 values). One block.
// ---------------------------------------------------------------
__global__ __launch_bounds__(256) void prep_kernel(const float* __restrict__ K4,
                                                   const float* __restrict__ datt,
                                                   float* __restrict__ ws) {
    int t = threadIdx.x;
    if (t < BNC) {
        const float* Kp = K4 + t * 16;   // 4x4 row-major, take top-left 3x3
        float a = Kp[0], b = Kp[1], c = Kp[2];
        float d = Kp[4], e = Kp[5], f = Kp[6];
        float g = Kp[8], h = Kp[9], i = Kp[10];
        float A  = e * i - f * h;
        float Bc = f * g - d * i;
        float Cc = d * h - e * g;
        float det = a * A + b * Bc + c * Cc;
        float inv = 1.0f / det;
        float* o = ws + OFF_KINV + t * 9;
        o[0] = A * inv;            o[1] = (c * h - b * i) * inv;  o[2] = (b * f - c * e) * inv;
        o[3] = Bc * inv;           o[4] = (a * i - c * g) * inv;  o[5] = (c * d - a * f) * inv;
        o[6] = Cc * inv;           o[7] = (b * g - a * h) * inv;  o[8] = (a * e - b * d) * inv;
    }
    if (t == 16) {   // depth softmax (48 values, trivially serial)
        float mx = -1e30f;
        for (int j = 0; j < DD; ++j) mx = fmaxf(mx, datt[j]);
        float e[DD]; float s = 0.f;
        for (int j = 0; j < DD; ++j) { e[j] = __expf(datt[j] - mx); s += e[j]; }
        float r = 1.0f / s;
        for (int j = 0; j < DD; ++j) ws[OFF_DA + j] = e[j] * r;
    }
}

// ---------------------------------------------------------------
// Kernel 2: per-pixel softmax over the N=6 views.
// ---------------------------------------------------------------
__global__ __launch_bounds__(256) void va_softmax_kernel(const float* __restrict__ va_in,
                                                         float* __restrict__ va_out) {
    int m = blockIdx.x * 256 + threadIdx.x;
    if (m >= HWSZ) return;
    float v[NN];
    float mx = -1e30f;
#pragma unroll
    for (int n = 0; n < NN; ++n) { v[n] = va_in[n * HWSZ + m]; mx = fmaxf(mx, v[n]); }
    float s = 0.f;
#pragma unroll
    for (int n = 0; n < NN; ++n) { v[n] = __expf(v[n] - mx); s += v[n]; }
    float r = 1.0f / s;
#pragma unroll
    for (int n = 0; n < NN; ++n) va_out[n * HWSZ + m] = v[n] * r;
}

// ---------------------------------------------------------------
// Kernel 3: pc writer (the 311 MB stream that bounds runtime).
// pc[((bn*3+c)*D+dd)*HW + p] = depths[dd] * (ki0*x + ki1*y + ki2),
// float4 stores, fully coalesced.
// ---------------------------------------------------------------
__global__ __launch_bounds__(256) void pc_kernel(const float* __restrict__ depths,
                                                 const float* __restrict__ ws,
                                                 float* __restrict__ pc) {
    const int PQ = HWSZ / 4;                       // 11264 quads per (bn,c)
    int idx = blockIdx.x * 256 + threadIdx.x;      // exact grid: BN*3*PQ threads
    int pq = idx % PQ;
    int c  = (idx / PQ) % 3;
    int bn = idx / (PQ * 3);
    int p  = pq * 4;
    int y  = p / WW, x = p % WW;                   // WW%4==0 -> same row for all 4
    const float* ki = ws + OFF_KINV + (bn * 3 + c) * 3;
    float k0 = ki[0];
    float r0 = k0 * (float)x + ki[1] * (float)y + ki[2];
    float r1 = r0 + k0, r2 = r0 + 2.f * k0, r3 = r0 + 3.f * k0;
    float* base = pc + (size_t)(bn * 3 + c) * DD * HWSZ + p;
#pragma unroll 8
    for (int dd = 0; dd < DD; ++dd) {
        float dep = depths[dd];
        float4 v = make_float4(dep * r0, dep * r1, dep * r2, dep * r3);
        *(float4*)(base + (size_t)dd * HWSZ) = v;
    }
}

// ---------------------------------------------------------------
// Kernel 4: bev_features[b,c,m] = sum_n va[n,m] * sum_d da[d]*pc[b,n,c, m*48+d]
// (the d-sum reads 48 *contiguous* floats thanks to the raw-reshape view).
// ---------------------------------------------------------------
__global__ __launch_bounds__(256) void bevfeat_kernel(const float* __restrict__ pc,
                                                      const float* __restrict__ ws,
                                                      float* __restrict__ bevf,
                                                      const float* __restrict__ va) {
    __shared__ float da_s[DD];
    if (threadIdx.x < DD) da_s[threadIdx.x] = ws[OFF_DA + threadIdx.x];
    __syncthreads();
    int idx = blockIdx.x * 256 + threadIdx.x;      // exact grid: BB*3*HWSZ
    int m = idx % HWSZ;
    int c = (idx / HWSZ) % 3;
    int b = idx / (3 * HWSZ);
    float acc = 0.f;
#pragma unroll
    for (int n = 0; n < NN; ++n) {
        float w = va[n * HWSZ + m];
        const float4* pp = (const float4*)(pc + (size_t)((b * NN + n) * 3 + c) * HWSZ * DD
                                              + (size_t)m * DD);
        float s = 0.f;
#pragma unroll
        for (int j = 0; j < DD / 4; ++j) {
            float4 v = pp[j];
            s += da_s[4 * j + 0] * v.x + da_s[4 * j + 1] * v.y
               + da_s[4 * j + 2] * v.z + da_s[4 * j + 3] * v.w;
        }
        acc += w * s;
    }
    bevf[idx] = acc;   // idx == (b*3+c)*HWSZ + m  -> (B,3,H,W) layout
}

// ---------------------------------------------------------------
// Kernel 5: conv1 (3 -> 64 channels, 3x3, SAME) direct: only 27 MACs/out.
// ---------------------------------------------------------------
__global__ __launch_bounds__(256) void conv1_kernel(const float* __restrict__ bevf,
                                                    const float* __restrict__ w1,
                                                    const float* __restrict__ b1,
                                                    float* __restrict__ hbuf) {
    int idx = blockIdx.x * 256 + threadIdx.x;      // exact grid: BB*BEVC*HWSZ
    int m = idx % HWSZ;
    int o = (idx / HWSZ) % BEVC;
    int b = idx / (BEVC * HWSZ);
    int y = m / WW, x = m % WW;
    float acc = b1[o];
#pragma unroll
    for (int i = 0; i < 3; ++i) {
        const float* src = bevf + (size_t)(b * 3 + i) * HWSZ;
        const float* wr  = w1 + (o * 3 + i) * 9;
#pragma unroll
        for (int ky = 0; ky < 3; ++ky) {
            int yy = y + ky - 1;
            if (yy < 0 || yy >= HH) continue;
#pragma unroll
            for (int kx = 0; kx < 3; ++kx) {
                int xx = x + kx - 1;
                if (xx < 0 || xx >= WW) continue;
                acc += wr[ky * 3 + kx] * src[yy * WW + xx];
            }
        }
    }
    hbuf[idx] = acc;
}

// ---------------------------------------------------------------
// Kernel 6: per-channel batch stats (sum, sumsq) over (B,H,W).
// One block per channel, LDS tree reduction — no atomics, so no
// workspace pre-zeroing is needed between graph replays.
// ---------------------------------------------------------------
__global__ __launch_bounds__(256) void bnstats_kernel(const float* __restrict__ x,
                                                      float* __restrict__ sums,
                                                      float* __restrict__ sumsq) {
    int o = blockIdx.x;
    float s = 0.f, q = 0.f;
    for (int t = threadIdx.x; t < BB * HWSZ; t += 256) {
        int b = t / HWSZ, m = t % HWSZ;
        float v = x[(size_t)(b * BEVC + o) * HWSZ + m];
        s += v; q += v * v;
    }
    __shared__ float ss[256], qq[256];
    ss[threadIdx.x] = s; qq[threadIdx.x] = q;
    __syncthreads();
    for (int st = 128; st > 0; st >>= 1) {
        if (threadIdx.x < st) {
            ss[threadIdx.x] += ss[threadIdx.x + st];
            qq[threadIdx.x] += qq[threadIdx.x + st];
        }
        __syncthreads();
    }
    if (threadIdx.x == 0) { sums[o] = ss[0]; sumsq[o] = qq[0]; }
}

// ---------------------------------------------------------------
// Kernel 7: BN (training-mode, biased var) + ReLU, elementwise.
// ---------------------------------------------------------------
__global__ __launch_bounds__(256) void bnrelu_kernel(const float* __restrict__ x,
                                                     const float* __restrict__ g,
                                                     const float* __restrict__ bb,
                                                     const float* __restrict__ sums,
                                                     const float* __restrict__ sumsq,
                                                     float* __restrict__ out) {
    int idx = blockIdx.x * 256 + threadIdx.x;      // exact grid: BB*BEVC*HWSZ
    int o = (idx / HWSZ) % BEVC;
    const float invN = 1.0f / (float)(BB * HWSZ);
    float mean = sums[o] * invN;
    float var  = sumsq[o] * invN - mean * mean;
    float sc   = g[o] * rsqrtf(var + 1e-5f);
    float v    = sc * (x[idx] - mean) + bb[o];
    out[idx] = fmaxf(v, 0.f);
}

// ---------------------------------------------------------------
// Kernel 8a: weight re-layout for the WMMA conv: tap-major K order.
//   wt[o*576 + (ky*3+kx)*64 + i] = c2w[((o*64+i)*3+ky)*3+kx]
// Makes the B-fragment (K, K+1) pair contiguous -> one b64 load.
// ---------------------------------------------------------------
__global__ __launch_bounds__(256) void wtr_kernel(const float* __restrict__ w2,
                                                  float* __restrict__ wt) {
    int idx = blockIdx.x * 256 + threadIdx.x;      // exact grid: 64*576 = 36864
    int i   = idx % 64;
    int tap = (idx / 64) % 9;
    int o   = idx / 576;
    wt[idx] = w2[(size_t)o * 576 + i * 9 + tap];
}

// ---------------------------------------------------------------
// Kernel 8b: conv2 (64 -> 64, 3x3, SAME) as implicit GEMM on the CDNA5
// matrix engine: V_WMMA_F32_16X16X4_F32 (f32 end-to-end).
//   M = 16 pixels of one row, N = 64 out channels (4 tiles of 16),
//   K = 576 in TAP-MAJOR order: kk = tap*64 + i  (A and B agree on the
//   permutation, so the GEMM is unchanged).
// Per tap (outer 9-iteration loop) the (ky,kx) offsets, the boundary
// mask and the base addresses are loop-invariant, so the hot loop is
// just: 2 strided b32 A-loads + 4 b64 B-loads + 4 WMMAs. Boundary is
// branch-free: clamped address load * {0,1} mask.
// A layout (16x4 f32): lanes 0-15 hold K=k,k+1; lanes 16-31 hold K=k+2,k+3.
// B layout (4x16 f32): same K split; lane%16 selects column (out channel).
// D layout (16x16 f32): vgpr r, lanes 0-15 -> M=r, lanes 16-31 -> M=8+r.
// ---------------------------------------------------------------
__global__ __launch_bounds__(256) void conv2_wmma_kernel(const float* __restrict__ hin,
                                                         const float* __restrict__ wt,
                                                         const float* __restrict__ b2,
                                                         float* __restrict__ y2) {
    const int wave = threadIdx.x >> 5;
    const int lane = threadIdx.x & 31;
    const int tile = blockIdx.x * 8 + wave;        // B*H*(W/16) = 5632 tiles
    const int b   = tile / (HH * 22);
    const int rem = tile % (HH * 22);
    const int y   = rem / 22;
    const int x0  = (rem % 22) * 16;
    const int laneM = lane & 15;
    const int hi    = lane >> 4;                   // 0: K=k,k+1  1: K=k+2,k+3
    const int px    = x0 + laneM;                  // this lane's A-row pixel

    v8f acc0, acc1, acc2, acc3;
    {
        float b0 = b2[ 0 + laneM], b1 = b2[16 + laneM];
        float b2v = b2[32 + laneM], b3 = b2[48 + laneM];
#pragma unroll
        for (int r = 0; r < 8; ++r) { acc0[r] = b0; acc1[r] = b1; acc2[r] = b2v; acc3[r] = b3; }
    }

    const float* hb = hin + (size_t)b * BEVC * HWSZ + (size_t)(2 * hi) * HWSZ;
    // B pointers, already offset by this half-wave's K offset (2*hi)
    const float* wp0 = wt + (size_t)( 0 + laneM) * 576 + 2 * hi;
    const float* wp1 = wt + (size_t)(16 + laneM) * 576 + 2 * hi;
    const float* wp2 = wt + (size_t)(32 + laneM) * 576 + 2 * hi;
    const float* wp3 = wt + (size_t)(48 + laneM) * 576 + 2 * hi;

#pragma unroll
    for (int tap = 0; tap < 9; ++tap) {
        const int ky = tap / 3, kx = tap % 3;      // compile-time (unrolled)
        const int yy = y + ky - 1;
        const int xx = px + kx - 1;
        const bool inb = (yy >= 0) & (yy < HH) & (xx >= 0) & (xx < WW);
        const float msk = inb ? 1.0f : 0.0f;
        const int yc = min(max(yy, 0), HH - 1);    // clamped, always-valid addr
        const int xc = min(max(xx, 0), WW - 1);
        const float* ap = hb + yc * WW + xc;       // A base for this tap
        const int kb = tap * 64;                   // K base for B
        for (int i = 0; i < 64; i += 4) {          // 16 WMMA steps per tap
            v2f af;
            af[0] = msk * ap[(size_t)(i    ) * HWSZ];
            af[1] = msk * ap[(size_t)(i + 1) * HWSZ];
            v2f bf0 = *(const v2f*)(wp0 + kb + i);
            v2f bf1 = *(const v2f*)(wp1 + kb + i);
            v2f bf2 = *(const v2f*)(wp2 + kb + i);
            v2f bf3 = *(const v2f*)(wp3 + kb + i);
            acc0 = __builtin_amdgcn_wmma_f32_16x16x4_f32(false, af, false, bf0, (short)0, acc0, false, false);
            acc1 = __builtin_amdgcn_wmma_f32_16x16x4_f32(false, af, false, bf1, (short)0, acc1, false, false);
            acc2 = __builtin_amdgcn_wmma_f32_16x16x4_f32(false, af, false, bf2, (short)0, acc2, false, false);
            acc3 = __builtin_amdgcn_wmma_f32_16x16x4_f32(false, af, false, bf3, (short)0, acc3, false, false);
        }
    }

    // --- store D: lane holds out-channel (tile*16 + laneM), rows M = r + 8*hi ---
    const int mbase = y * WW + x0;
#pragma unroll
    for (int r = 0; r < 8; ++r) {
        int M = r + 8 * hi;
        size_t pm = (size_t)mbase + M;
        y2[((size_t)(b * BEVC) +  0 + laneM) * HWSZ + pm] = acc0[r];
        y2[((size_t)(b * BEVC) + 16 + laneM) * HWSZ + pm] = acc1[r];
        y2[((size_t)(b * BEVC) + 32 + laneM) * HWSZ + pm] = acc2[r];
        y2[((size_t)(b * BEVC) + 48 + laneM) * HWSZ + pm] = acc3[r];
    }
}

// ---------------------------------------------------------------
extern "C" void kernel_launch(void* const* d_in, const int* in_sizes, int n_in,
                              void* d_out, int out_size, void* d_ws, size_t ws_size,
                              hipStream_t stream) {
    // setup_inputs order:
    // 0 image_features (unused) 1 intrinsics 2 extrinsics (unused)
    // 3 view_attention 4 depth_attention 5 c1w 6 c1b 7 bn1_g 8 bn1_b
    // 9 c2w 10 c2b 11 bn2_g 12 bn2_b 13 depths 14 xy1 (recomputed analytically)
    const float* intr  = (const float*)d_in[1];
    const float* va_in = (const float*)d_in[3];
    const float* datt  = (const float*)d_in[4];
    const float* c1w   = (const float*)d_in[5];
    const float* c1b   = (const float*)d_in[6];
    const float* bn1g  = (const float*)d_in[7];
    const float* bn1b  = (const float*)d_in[8];
    const float* c2w   = (const float*)d_in[9];
    const float* c2b   = (const float*)d_in[10];
    const float* bn2g  = (const float*)d_in[11];
    const float* bn2b  = (const float*)d_in[12];
    const float* depths = (const float*)d_in[13];

    float* out = (float*)d_out;           // [pc (77,856,768) | bev (5,767,168)]
    float* ws  = (float*)d_ws;            // needs ~48.5 MB

    const int ELEM = BB * BEVC * HWSZ;    // 5,767,168 (divisible by 256)

    prep_kernel<<<1, 256, 0, stream>>>(intr, datt, ws);
    va_softmax_kernel<<<(HWSZ + 255) / 256, 256, 0, stream>>>(va_in, ws + OFF_VA);
    wtr_kernel<<<(BEVC * 576) / 256, 256, 0, stream>>>(c2w, ws + OFF_WT);

    // pc: BN*3*(HW/4) threads = 405,504 -> 1584 blocks (exact)
    pc_kernel<<<(BNC * 3 * (HWSZ / 4)) / 256, 256, 0, stream>>>(depths, ws, out);

    // bev features: BB*3*HW = 270,336 -> 1056 blocks (exact)
    bevfeat_kernel<<<(BB * 3 * HWSZ) / 256, 256, 0, stream>>>(out, ws, ws + OFF_BEVF, ws + OFF_VA);

    conv1_kernel<<<ELEM / 256, 256, 0, stream>>>(ws + OFF_BEVF, c1w, c1b, ws + OFF_H);
    bnstats_kernel<<<BEVC, 256, 0, stream>>>(ws + OFF_H, ws + OFF_S1, ws + OFF_Q1);
    bnrelu_kernel<<<ELEM / 256, 256, 0, stream>>>(ws + OFF_H, bn1g, bn1b,
                                                  ws + OFF_S1, ws + OFF_Q1, ws + OFF_H); // in place

    conv2_wmma_kernel<<<(BB * HH * 22) / 8, 256, 0, stream>>>(ws + OFF_H, ws + OFF_WT, c2b, ws + OFF_Y2);
    bnstats_kernel<<<BEVC, 256, 0, stream>>>(ws + OFF_Y2, ws + OFF_S2, ws + OFF_Q2);
    bnrelu_kernel<<<ELEM / 256, 256, 0, stream>>>(ws + OFF_Y2, bn2g, bn2b,
                                                  ws + OFF_S2, ws + OFF_Q2, out + PCSIZE);
}